// LSTMPredictor_75007308857624
// MI455X (gfx1250) — compile-verified
//
#include <hip/hip_runtime.h>

// LSTM (B=4096, T=512, IN=3, H=64) on CDNA5 using fp32 WMMA 16x16x4.
// One wave = 16 batch rows for all 512 timesteps; weights + h live in LDS.
// Weights stored K-pair-packed (float2): every B fragment is one ds_load_b64,
// and gate-adjacent fragments merge into ds_load_2addr_b64.
// Bias folded in as K-row 67 (A supplies 1.0); the x/bias K-step runs FIRST
// with a literal-zero C so the accumulators are born from WMMA (inline 0 SRC2).

typedef __attribute__((ext_vector_type(2))) float v2f;
typedef __attribute__((ext_vector_type(8))) float v8f;

#define T_LEN 512
#define WTP_STRIDE 272  // float2 per K-pair row: lane-half offset = 32 banks
#define H_STRIDE 68     // float  per h row: bank-conflict free b64 A loads

static __device__ __forceinline__ v8f wmma4(v2f a, v2f b, v8f c) {
  // D = A(16x4 f32) * B(4x16 f32) + C(16x16 f32)
  return __builtin_amdgcn_wmma_f32_16x16x4_f32(false, a, false, b, (short)0, c,
                                               false, false);
}

static __device__ __forceinline__ float sigm(float x) {
  return 1.0f / (1.0f + __expf(-x));
}
static __device__ __forceinline__ float tanh_(float x) {
  return 1.0f - 2.0f / (1.0f + __expf(2.0f * x));
}

__global__ __launch_bounds__(128) void lstm_wmma_kernel(
    const float* __restrict__ x, const float* __restrict__ W_ih,
    const float* __restrict__ W_hh, const float* __restrict__ b_ih,
    const float* __restrict__ b_hh, const float* __restrict__ W_out,
    const float* __restrict__ b_out, float* __restrict__ out) {
  // Logical K rows 0..63 = W_hh^T, 64..66 = W_ih^T, 67 = (b_ih+b_hh).
  // Packed: WtP[(k/2)*WTP_STRIDE + n] holds float2 {row k, row k+1}, k even.
  __shared__ float WtP[34 * WTP_STRIDE * 2];
  __shared__ float WtOutP[32 * 16 * 2];    // {W_out[n][2p], W_out[n][2p+1]}
  __shared__ float hS[4 * 16 * H_STRIDE];  // per-wave h[16][68]

  const int tid  = threadIdx.x;
  const int wave = tid >> 5;
  const int lane = tid & 31;
  const int col  = lane & 15;  // N / A-row index inside fragments
  const int hi   = lane >> 4;  // lane half: selects K+2 / M+8
  const int bbase = blockIdx.x * 64 + wave * 16;

  // ---- one-time LDS init ----
  for (int idx = tid; idx < 68 * 256; idx += 128) {
    int k = idx >> 8, n = idx & 255;
    float v;
    if (k < 64)      v = W_hh[n * 64 + k];
    else if (k < 67) v = W_ih[n * 3 + (k - 64)];
    else             v = b_ih[n] + b_hh[n];  // bias row (A supplies 1.0)
    WtP[((k >> 1) * WTP_STRIDE + n) * 2 + (k & 1)] = v;
  }
  for (int idx = tid; idx < 64 * 16; idx += 128) {
    int k = idx >> 4, n = idx & 15;
    WtOutP[((k >> 1) * 16 + n) * 2 + (k & 1)] =
        (n < 3) ? W_out[n * 64 + k] : 0.0f;
  }
  for (int idx = tid; idx < 4 * 16 * H_STRIDE; idx += 128) hS[idx] = 0.0f;
  __syncthreads();

  float* hW = &hS[wave * (16 * H_STRIDE)];
  const float* xrow = x + (size_t)(bbase + col) * (T_LEN * 3);
  const float bO = (col < 3) ? b_out[col] : 0.0f;
  const v8f vzero = {0, 0, 0, 0, 0, 0, 0, 0};

  v8f cs0 = vzero, cs1 = vzero, cs2 = vzero, cs3 = vzero;

  for (int t = 0; t < T_LEN; ++t) {
    // A-fragment for the x/bias K-step (rows 64..67): [x0,x1 | x2,1.0]
    const float* xp = xrow + t * 3;
    v2f ax;
    if (hi == 0) { ax[0] = xp[0]; ax[1] = xp[1]; }
    else         { ax[0] = xp[2]; ax[1] = 1.0f; }

    v8f hn0, hn1, hn2, hn3;

#define GATE_BLOCK(J, CS, HN)                                                 \
    {                                                                         \
      const int nb = (J) * 16 + col;                                          \
      v8f ai, af, ag, ao;                                                     \
      { /* x @ W_ih^T + bias first, C = inline 0 (pairs 32 / 33) */           \
        const float* wp = &WtP[((32 + hi) * WTP_STRIDE + nb) * 2];            \
        v2f b0 = *(const v2f*)(wp);                                           \
        v2f b1 = *(const v2f*)(wp + 128);                                     \
        v2f b2 = *(const v2f*)(wp + 256);                                     \
        v2f b3 = *(const v2f*)(wp + 384);                                     \
        ai = wmma4(ax, b0, vzero);                                            \
        af = wmma4(ax, b1, vzero);                                            \
        ag = wmma4(ax, b2, vzero);                                            \
        ao = wmma4(ax, b3, vzero);                                            \
      }                                                                       \
      _Pragma("unroll 2")                                                     \
      for (int ks = 0; ks < 16; ++ks) {                                       \
        v2f a = *(const v2f*)&hW[col * H_STRIDE + ks * 4 + hi * 2];           \
        const float* wp = &WtP[((ks * 2 + hi) * WTP_STRIDE + nb) * 2];        \
        v2f b0 = *(const v2f*)(wp);                                           \
        v2f b1 = *(const v2f*)(wp + 128);                                     \
        v2f b2 = *(const v2f*)(wp + 256);                                     \
        v2f b3 = *(const v2f*)(wp + 384);                                     \
        ai = wmma4(a, b0, ai);                                                \
        af = wmma4(a, b1, af);                                                \
        ag = wmma4(a, b2, ag);                                                \
        ao = wmma4(a, b3, ao);                                                \
      }                                                                       \
      _Pragma("unroll")                                                       \
      for (int r = 0; r < 8; ++r) {                                           \
        float iv = sigm(ai[r]), fv = sigm(af[r]);                             \
        float gv = tanh_(ag[r]), ov = sigm(ao[r]);                            \
        float cv = fv * CS[r] + iv * gv;                                      \
        CS[r] = cv;                                                           \
        HN[r] = ov * tanh_(cv);                                               \
      }                                                                       \
    }

    GATE_BLOCK(0, cs0, hn0)
    GATE_BLOCK(1, cs1, hn1)
    GATE_BLOCK(2, cs2, hn2)
    GATE_BLOCK(3, cs3, hn3)
#undef GATE_BLOCK

    // publish h_new to wave-private LDS (D-layout row = r + hi*8)
    {
      float* hp = &hW[hi * 8 * H_STRIDE + col];
#pragma unroll
      for (int r = 0; r < 8; ++r) {
        hp[r * H_STRIDE +  0] = hn0[r];
        hp[r * H_STRIDE + 16] = hn1[r];
        hp[r * H_STRIDE + 32] = hn2[r];
        hp[r * H_STRIDE + 48] = hn3[r];
      }
    }

    // out = h_new @ W_out^T + b_out   (N=16 tile, only cols 0..2 valid)
    v8f aco = {bO, bO, bO, bO, bO, bO, bO, bO};
#pragma unroll 2
    for (int ks = 0; ks < 16; ++ks) {
      v2f a = *(const v2f*)&hW[col * H_STRIDE + ks * 4 + hi * 2];
      v2f b = *(const v2f*)&WtOutP[((ks * 2 + hi) * 16 + col) * 2];
      aco = wmma4(a, b, aco);
    }
    if (col < 3) {
#pragma unroll
      for (int r = 0; r < 8; ++r) {
        size_t bidx = (size_t)(bbase + hi * 8 + r);
        out[(bidx * T_LEN + t) * 3 + col] = aco[r];
      }
    }
  }
}

extern "C" void kernel_launch(void* const* d_in, const int* in_sizes, int n_in,
                              void* d_out, int out_size, void* d_ws,
                              size_t ws_size, hipStream_t stream) {
  const float* x     = (const float*)d_in[0];
  const float* W_ih  = (const float*)d_in[1];
  const float* W_hh  = (const float*)d_in[2];
  const float* b_ih  = (const float*)d_in[3];
  const float* b_hh  = (const float*)d_in[4];
  const float* W_out = (const float*)d_in[5];
  const float* b_out = (const float*)d_in[6];
  float* out = (float*)d_out;

  dim3 grid(4096 / 64);  // 64 workgroups x 4 waves x 16 batch rows = 4096
  dim3 block(128);
  lstm_wmma_kernel<<<grid, block, 0, stream>>>(x, W_ih, W_hh, b_ih, b_hh,
                                               W_out, b_out, out);
}